// GCNBackbone_24386824306774
// MI455X (gfx1250) — compile-verified
//
#include <hip/hip_runtime.h>

#define GCN_N 100000
#define GCN_D 128

typedef float v2f __attribute__((ext_vector_type(2)));
typedef float v8f __attribute__((ext_vector_type(8)));

// ---------------- utility fills ----------------
__global__ __launch_bounds__(256) void gcn_fill_zero(float* __restrict__ p, long long n) {
    long long i = (long long)blockIdx.x * blockDim.x + threadIdx.x;
    if (i < n) p[i] = 0.0f;
}

// ---------------- degree accumulation ----------------
__global__ __launch_bounds__(256) void gcn_deg_accum(const int* __restrict__ dst,
                                                     float* __restrict__ deg, int ne) {
    int i = blockIdx.x * blockDim.x + threadIdx.x;
    if (i < ne) atomicAdd(&deg[dst[i]], 1.0f);
}

__global__ __launch_bounds__(256) void gcn_finalize_dinv(float* __restrict__ dinv, int n) {
    int i = blockIdx.x * blockDim.x + threadIdx.x;
    if (i < n) dinv[i] = rsqrtf(dinv[i] + 1.0f);   // +1 for self loop; deg>=1 always
}

// ---------------- fp32 WMMA GEMM: H[N,128] = X[N,128] @ W[128,128] ----------------
// One wave computes one 16x16 output tile via V_WMMA_F32_16X16X4_F32, K=128 in 32 steps.
// A (16x4 f32): lanes 0-15 hold {K=0,K=1}, lanes 16-31 hold {K=2,K=3} of row M=lane&15.
// B (4x16 f32): same striping with N=lane&15.
// C/D (16x16 f32, 8 VGPRs): row = i + 8*(lane>=16), col = lane&15.
__global__ __launch_bounds__(256) void gcn_gemm_wmma(const float* __restrict__ X,
                                                     const float* __restrict__ W,
                                                     float* __restrict__ H) {
    const int lane     = threadIdx.x & 31;
    const int wave     = blockIdx.x * 8 + (threadIdx.x >> 5);
    const int col_tile = wave & 7;       // 8 tiles of 16 columns cover D=128
    const int row_tile = wave >> 3;      // N/16 = 6250 row tiles (exact)
    const int m        = lane & 15;
    const int half     = lane >> 4;      // 0 or 1
    const int koff     = half << 1;      // 0 or 2

    const float* xrow = X + (size_t)(row_tile * 16 + m) * GCN_D;
    const int    ncol = col_tile * 16 + m;

    v8f acc = {};
#pragma unroll
    for (int k = 0; k < GCN_D; k += 4) {
        v2f a = *(const v2f*)(xrow + k + koff);          // 8B aligned (k+koff even)
        v2f b;
        b.x = W[(size_t)(k + koff)     * GCN_D + ncol];
        b.y = W[(size_t)(k + koff + 1) * GCN_D + ncol];
        acc = __builtin_amdgcn_wmma_f32_16x16x4_f32(
                  false, a, false, b, (short)0, acc, false, false);
    }

    const int mbase = row_tile * 16 + half * 8;
    float* hp = H + (size_t)mbase * GCN_D + col_tile * 16 + m;
#pragma unroll
    for (int i = 0; i < 8; ++i) hp[(size_t)i * GCN_D] = acc[i];
}

// ---------------- edge scatter: acc[dst] += h[src] * dinv[src]*dinv[dst] ----------------
// One wave per edge; src/dst indices are wave-uniform (scalar loads); each lane moves float4.
__global__ __launch_bounds__(256) void gcn_scatter(const int* __restrict__ src,
                                                   const int* __restrict__ dst,
                                                   const float* __restrict__ h,
                                                   const float* __restrict__ dinv,
                                                   float* __restrict__ acc, int ne) {
    int warp = (int)((blockIdx.x * blockDim.x + threadIdx.x) >> 5);
    int lane = threadIdx.x & 31;
    if (warp >= ne) return;                 // wave-uniform branch
    int s = src[warp];
    int d = dst[warp];
    float nrm = dinv[s] * dinv[d];
    float4 v = ((const float4*)(h + (size_t)s * GCN_D))[lane];
    float* o = acc + (size_t)d * GCN_D + lane * 4;
    atomicAdd(o + 0, v.x * nrm);
    atomicAdd(o + 1, v.y * nrm);
    atomicAdd(o + 2, v.z * nrm);
    atomicAdd(o + 3, v.w * nrm);
}

// ---------------- epilogue: out = relu(acc + h*dinv^2 (self loop) + b) ----------------
__global__ __launch_bounds__(256) void gcn_epilogue(const float* __restrict__ acc,
                                                    const float* __restrict__ h,
                                                    const float* __restrict__ dinv,
                                                    const float* __restrict__ b,
                                                    float* __restrict__ out, long long n) {
    long long i = (long long)blockIdx.x * blockDim.x + threadIdx.x;
    if (i >= n) return;
    int node = (int)(i >> 7);
    int j    = (int)(i & 127);
    float di = dinv[node];
    float v  = acc[i] + h[i] * di * di + b[j];
    out[i] = v > 0.0f ? v : 0.0f;
}

extern "C" void kernel_launch(void* const* d_in, const int* in_sizes, int n_in,
                              void* d_out, int out_size, void* d_ws, size_t ws_size,
                              hipStream_t stream) {
    const float* x  = (const float*)d_in[0];
    const float* W1 = (const float*)d_in[1];
    const float* b1 = (const float*)d_in[2];
    const float* W2 = (const float*)d_in[3];
    const float* b2 = (const float*)d_in[4];
    const float* W3 = (const float*)d_in[5];
    const float* b3 = (const float*)d_in[6];
    const int*   eidx = (const int*)d_in[7];
    const int ne = in_sizes[7] / 2;               // 1,600,000
    const int* src = eidx;
    const int* dst = eidx + ne;

    // workspace layout (tight pack): dinv | h | acc | buf
    char* ws = (char*)d_ws;
    size_t off = 0;
    float* dinv = (float*)(ws + off); off += (((size_t)GCN_N * 4) + 255) & ~(size_t)255;
    float* h    = (float*)(ws + off); off += (size_t)GCN_N * GCN_D * 4;
    float* acc  = (float*)(ws + off); off += (size_t)GCN_N * GCN_D * 4;
    float* buf  = (float*)(ws + off);

    const long long feat_elems = (long long)GCN_N * GCN_D;   // 12.8M
    const int blk = 256;
    const int g_node  = (GCN_N + blk - 1) / blk;
    const int g_edge  = (ne + blk - 1) / blk;
    const long long g_feat_ll = (feat_elems + blk - 1) / blk;
    const int g_feat  = (int)g_feat_ll;
    const int g_gemm  = GCN_N / 16;                           // 6250 blocks x 8 waves
    const int g_scat  = (int)(((long long)ne * 32 + blk - 1) / blk);

    // --- degrees -> dinv ---
    gcn_fill_zero<<<g_node, blk, 0, stream>>>(dinv, GCN_N);
    gcn_deg_accum<<<g_edge, blk, 0, stream>>>(dst, dinv, ne);
    gcn_finalize_dinv<<<g_node, blk, 0, stream>>>(dinv, GCN_N);

    // --- layer 1 ---
    gcn_gemm_wmma<<<g_gemm, blk, 0, stream>>>(x, W1, h);
    gcn_fill_zero<<<g_feat, blk, 0, stream>>>(acc, feat_elems);
    gcn_scatter<<<g_scat, blk, 0, stream>>>(src, dst, h, dinv, acc, ne);
    gcn_epilogue<<<g_feat, blk, 0, stream>>>(acc, h, dinv, b1, buf, feat_elems);

    // --- layer 2 ---
    gcn_gemm_wmma<<<g_gemm, blk, 0, stream>>>(buf, W2, h);
    gcn_fill_zero<<<g_feat, blk, 0, stream>>>(acc, feat_elems);
    gcn_scatter<<<g_scat, blk, 0, stream>>>(src, dst, h, dinv, acc, ne);
    gcn_epilogue<<<g_feat, blk, 0, stream>>>(acc, h, dinv, b2, buf, feat_elems);

    // --- layer 3 ---
    gcn_gemm_wmma<<<g_gemm, blk, 0, stream>>>(buf, W3, h);
    gcn_fill_zero<<<g_feat, blk, 0, stream>>>(acc, feat_elems);
    gcn_scatter<<<g_scat, blk, 0, stream>>>(src, dst, h, dinv, acc, ne);
    gcn_epilogue<<<g_feat, blk, 0, stream>>>(acc, h, dinv, b3, (float*)d_out, feat_elems);
}